// GraphWasserstein1Loss_E_79731772883224
// MI455X (gfx1250) — compile-verified
//
#include <hip/hip_runtime.h>
#include <hip/hip_bf16.h>
#include <math.h>

// ---------------------------------------------------------------------------
// GraphWasserstein loss on MI455X (gfx1250, wave32, WMMA).
//
// Pipeline (all on `stream`, deterministic, scratch in d_ws):
//   1) edge weights  w_e = exp(-||x_i - x_j||^2)          (wave-per-edge, shfl reduce)
//   2) scatter       Yacc[i,:] += w_e * P[:,j],  sym      (f32 atomics, L2 resident)
//   3) M = L @ P^T   = (deg-1)*P^T - Yacc                 (implicit Laplacian)
//   4) L_adj = P @ M + 1e-5 I                             (WMMA f32 GEMM, 512x8192x512)
//   5) S1 = inv(L_adj)    : Newton-Schulz  X <- X(2I - A X)
//   6) tr(S_) = sqrt(||B||_F) * tr(Y_inf),  B = S0 @ S1,  coupled Newton-Schulz sqrt
//   7) out = tr(S1) + tr(S0) - 2 tr(S_)
//
// GEMM: 4 waves/block, 64x64 block tile, 32x32 (2x2 fragments) per wave ->
// each A/B fragment feeds 2 WMMAs. Staging uses GLOBAL_LOAD_ASYNC_TO_LDS_B128
// (ASYNCcnt) with double buffering + s_wait_asynccnt when available.
// ---------------------------------------------------------------------------

typedef __attribute__((ext_vector_type(2))) float v2f;
typedef __attribute__((ext_vector_type(8))) float v8f;
typedef int v4i_vs __attribute__((vector_size(16)));  // matches builtin's param type

#define TILE_K 64
#define BM 64
#define BN 64
#define LDP 68  // padded row stride (68 mod 64 banks = 4 -> conflict-free columns)

#if defined(__has_builtin)
#if __has_builtin(__builtin_amdgcn_global_load_async_to_lds_b128) && \
    __has_builtin(__builtin_amdgcn_s_wait_asynccnt)
#define USE_ASYNC_LDS 1
#endif
#endif
#ifndef USE_ASYNC_LDS
#define USE_ASYNC_LDS 0
#endif

// ------------------------- WMMA f32 GEMM: C = alpha*(A@B) + gamma*I ---------
// A: [Mdim x Kdim] row-major, B: [Kdim x Ndim] row-major, C: [Mdim x Ndim].
// Requires Mdim,Ndim % 64 == 0 and Kdim % 64 == 0 (holds: 512, 8192).
__global__ __launch_bounds__(128)
void gemm_wmma_f32(float* __restrict__ C, const float* __restrict__ A,
                   const float* __restrict__ B, int Mdim, int Ndim, int Kdim,
                   float alpha, float gamma) {
#if USE_ASYNC_LDS
  __shared__ alignas(16) float As[2][BM][LDP];
  __shared__ alignas(16) float Bs[2][TILE_K][LDP];
#else
  __shared__ alignas(16) float As[1][BM][LDP];
  __shared__ alignas(16) float Bs[1][TILE_K][LDP];
#endif

  const int tid   = threadIdx.x;
  const int lane  = tid & 31;
  const int wave  = tid >> 5;          // 0..3
  const int wr    = wave >> 1;         // wave row (2x2 wave grid)
  const int wc    = wave & 1;          // wave col
  const int row16 = lane & 15;
  const int half  = lane >> 4;         // 0: lanes 0-15, 1: lanes 16-31
  const int bRow  = blockIdx.y * BM;
  const int bCol  = blockIdx.x * BN;

  v8f acc00 = {0.f, 0.f, 0.f, 0.f, 0.f, 0.f, 0.f, 0.f};
  v8f acc01 = acc00, acc10 = acc00, acc11 = acc00;

  const int aRow0 = wr * 32 + row16;        // fragment row groups within block tile
  const int aRow1 = wr * 32 + 16 + row16;
  const int bCol0 = wc * 32 + row16;
  const int bCol1 = wc * 32 + 16 + row16;

  const int nT = Kdim / TILE_K;

#if USE_ASYNC_LDS
  // ---- async (ASYNCcnt-tracked) direct global->LDS staging, double buffered
  auto issue_tile = [&](int k0, int buf) {
    // A tile: 64 rows x 64 cols = 1024 float4; 128 threads x 8 b128 ops
#pragma unroll
    for (int i = 0; i < 8; ++i) {
      const int f = tid + i * 128;
      const int r = f >> 4, c4 = f & 15;
      __builtin_amdgcn_global_load_async_to_lds_b128(
          (v4i_vs*)(&A[(size_t)(bRow + r) * Kdim + k0 + c4 * 4]),
          (v4i_vs*)(&As[buf][r][c4 * 4]), 0, 0);
    }
    // B tile: 64 rows x 64 cols
#pragma unroll
    for (int i = 0; i < 8; ++i) {
      const int f = tid + i * 128;
      const int r = f >> 4, c4 = f & 15;
      __builtin_amdgcn_global_load_async_to_lds_b128(
          (v4i_vs*)(&B[(size_t)(k0 + r) * Ndim + bCol + c4 * 4]),
          (v4i_vs*)(&Bs[buf][r][c4 * 4]), 0, 0);
    }
  };
  issue_tile(0, 0);
#endif

  for (int t = 0; t < nT; ++t) {
#if USE_ASYNC_LDS
    const int cur = t & 1;
    if (t + 1 < nT) {
      issue_tile((t + 1) * TILE_K, cur ^ 1);
      // 16 async ops just issued for next tile; in-order completion =>
      // leaving 16 outstanding means the current tile's 16 are done.
      __builtin_amdgcn_s_wait_asynccnt(16);
    } else {
      __builtin_amdgcn_s_wait_asynccnt(0);
    }
    __syncthreads();
#else
    const int cur = 0;
    const int k0 = t * TILE_K;
    // synchronous float4 staging fallback
#pragma unroll
    for (int i = 0; i < 8; ++i) {
      const int f = tid + i * 128;
      const int r = f >> 4, c4 = f & 15;
      *reinterpret_cast<float4*>(&As[0][r][c4 * 4]) =
          *reinterpret_cast<const float4*>(&A[(size_t)(bRow + r) * Kdim + k0 + c4 * 4]);
      *reinterpret_cast<float4*>(&Bs[0][r][c4 * 4]) =
          *reinterpret_cast<const float4*>(&B[(size_t)(k0 + r) * Ndim + bCol + c4 * 4]);
    }
    if (k0 + TILE_K < Kdim) {
      __builtin_prefetch(&A[(size_t)(bRow + (tid >> 1)) * Kdim + k0 + TILE_K], 0, 1);
      __builtin_prefetch(&B[(size_t)(k0 + TILE_K + (tid >> 1)) * Ndim + bCol], 0, 1);
    }
    __syncthreads();
#endif

    // ---- 16 k-steps x 4 WMMAs: V_WMMA_F32_16X16X4_F32
#pragma unroll
    for (int kk = 0; kk < TILE_K; kk += 4) {
      // A fragment, ISA 16x4 f32 layout: lanes 0-15 hold K=0,1; lanes 16-31 K=2,3
      v2f a0, a1, b0, b1;
      a0.x = As[cur][aRow0][kk + half * 2 + 0];
      a0.y = As[cur][aRow0][kk + half * 2 + 1];
      a1.x = As[cur][aRow1][kk + half * 2 + 0];
      a1.y = As[cur][aRow1][kk + half * 2 + 1];
      b0.x = Bs[cur][kk + half * 2 + 0][bCol0];
      b0.y = Bs[cur][kk + half * 2 + 1][bCol0];
      b1.x = Bs[cur][kk + half * 2 + 0][bCol1];
      b1.y = Bs[cur][kk + half * 2 + 1][bCol1];
      acc00 = __builtin_amdgcn_wmma_f32_16x16x4_f32(false, a0, false, b0, (short)0,
                                                    acc00, false, false);
      acc01 = __builtin_amdgcn_wmma_f32_16x16x4_f32(false, a0, false, b1, (short)0,
                                                    acc01, false, false);
      acc10 = __builtin_amdgcn_wmma_f32_16x16x4_f32(false, a1, false, b0, (short)0,
                                                    acc10, false, false);
      acc11 = __builtin_amdgcn_wmma_f32_16x16x4_f32(false, a1, false, b1, (short)0,
                                                    acc11, false, false);
    }
    __syncthreads();
  }

  // ---- epilogue: C/D 16x16 f32 layout (VGPR g -> M = g + 8*half, N = row16)
#pragma unroll
  for (int g = 0; g < 8; ++g) {
    const int r0 = bRow + wr * 32 + g + 8 * half;
    const int c0 = bCol + wc * 32 + row16;
    const int r1 = r0 + 16;
    const int c1 = c0 + 16;
    float v;
    v = alpha * acc00[g]; if (r0 == c0) v += gamma; C[(size_t)r0 * Ndim + c0] = v;
    v = alpha * acc01[g]; if (r0 == c1) v += gamma; C[(size_t)r0 * Ndim + c1] = v;
    v = alpha * acc10[g]; if (r1 == c0) v += gamma; C[(size_t)r1 * Ndim + c0] = v;
    v = alpha * acc11[g]; if (r1 == c1) v += gamma; C[(size_t)r1 * Ndim + c1] = v;
  }
}

// ------------------------- edge weights: one wave per edge ------------------
__global__ __launch_bounds__(256)
void edge_weight_kernel(const float* __restrict__ x, const int* __restrict__ ei,
                        float* __restrict__ w, int E) {
  const int lane = threadIdx.x & 31;
  const int wib  = threadIdx.x >> 5;
  const long e   = (long)blockIdx.x * 8 + wib;
  if (e >= E) return;
  const int i = ei[e];
  const int j = ei[(size_t)E + e];
  // D = 128 = 32 lanes x float4
  const float4 a = *reinterpret_cast<const float4*>(&x[(size_t)i * 128 + lane * 4]);
  const float4 b = *reinterpret_cast<const float4*>(&x[(size_t)j * 128 + lane * 4]);
  const float dx = a.x - b.x, dy = a.y - b.y, dz = a.z - b.z, dw = a.w - b.w;
  float d = dx * dx + dy * dy + dz * dz + dw * dw;
#pragma unroll
  for (int m = 16; m > 0; m >>= 1) d += __shfl_xor(d, m, 32);
  if (lane == 0) w[e] = expf(-d);
}

// ------------------------- init: Yacc=0, deg=1, scalars=0 -------------------
__global__ void init_kernel(float* __restrict__ Yacc, float* __restrict__ deg,
                            float* __restrict__ scal, long VK, int V) {
  long idx = (long)blockIdx.x * blockDim.x + threadIdx.x;
  long stride = (long)gridDim.x * blockDim.x;
  for (long t = idx; t < VK; t += stride) {
    Yacc[t] = 0.f;
    if (t < V) deg[t] = 1.f;
    if (t < 64) scal[t] = 0.f;
  }
}

// ------------------------- scatter: Yacc and deg ----------------------------
__global__ __launch_bounds__(512)
void scatter_kernel(const float* __restrict__ w, const int* __restrict__ ei,
                    const float* __restrict__ pmat, float* __restrict__ Yacc,
                    float* __restrict__ deg, int E, int V, int K) {
  const int e = blockIdx.x;
  const int c = threadIdx.x;
  const int i = ei[e];
  const int j = ei[(size_t)E + e];
  const float we = w[e];
  if (c == 0) atomicAdd(&deg[i], we);
  if (c == 1) atomicAdd(&deg[j], we);
  const float pj = pmat[(size_t)c * V + j];  // P^T[j,c]
  const float pi = pmat[(size_t)c * V + i];
  atomicAdd(&Yacc[(size_t)i * K + c], we * pj);
  atomicAdd(&Yacc[(size_t)j * K + c], we * pi);
}

// ------------------------- M = (deg-1) .* P^T - Yacc ------------------------
__global__ void form_M_kernel(float* __restrict__ M, const float* __restrict__ Yacc,
                              const float* __restrict__ deg,
                              const float* __restrict__ pmat, int V, int K) {
  long idx = (long)blockIdx.x * blockDim.x + threadIdx.x;
  long total = (long)V * K;
  long stride = (long)gridDim.x * blockDim.x;
  for (long t = idx; t < total; t += stride) {
    const int v = (int)(t / K);
    const int c = (int)(t % K);
    M[t] = (deg[v] - 1.f) * pmat[(size_t)c * V + v] - Yacc[t];
  }
}

// ------------------------- trace (single block, n <= 512) -------------------
__global__ __launch_bounds__(512)
void trace_kernel(const float* __restrict__ A, int n, int ld,
                  float* __restrict__ slot) {
  __shared__ float red[512];
  const int t = threadIdx.x;
  red[t] = (t < n) ? A[(size_t)t * ld + t] : 0.f;
  __syncthreads();
  for (int s = 256; s > 0; s >>= 1) {
    if (t < s) red[t] += red[t + s];
    __syncthreads();
  }
  if (t == 0) *slot = red[0];
}

// ------------------------- max row abs-sum (||A||_inf, n <= 512) ------------
__global__ __launch_bounds__(512)
void rowabsmax_kernel(const float* __restrict__ A, int n, float* __restrict__ slot) {
  __shared__ float red[512];
  const int t = threadIdx.x;
  float rs = 0.f;
  if (t < n)
    for (int c = 0; c < n; ++c) rs += fabsf(A[(size_t)t * n + c]);
  red[t] = rs;
  __syncthreads();
  for (int s = 256; s > 0; s >>= 1) {
    if (t < s) red[t] = fmaxf(red[t], red[t + s]);
    __syncthreads();
  }
  if (t == 0) *slot = red[0];
}

// ------------------------- Frobenius^2 (single block) -----------------------
__global__ __launch_bounds__(1024)
void fro2_kernel(const float* __restrict__ A, long n, float* __restrict__ slot) {
  __shared__ float red[1024];
  float s = 0.f;
  for (long t = threadIdx.x; t < n; t += 1024) {
    const float v = A[t];
    s += v * v;
  }
  red[threadIdx.x] = s;
  __syncthreads();
  for (int s2 = 512; s2 > 0; s2 >>= 1) {
    if ((int)threadIdx.x < s2) red[threadIdx.x] += red[threadIdx.x + s2];
    __syncthreads();
  }
  if (threadIdx.x == 0) *slot = red[0];
}

// ------------------------- elementwise scale by device scalar ---------------
__global__ void scale_kernel(float* __restrict__ dst, const float* __restrict__ src,
                             const float* __restrict__ s, long n) {
  long idx = (long)blockIdx.x * blockDim.x + threadIdx.x;
  long stride = (long)gridDim.x * blockDim.x;
  const float sc = *s;
  for (long t = idx; t < n; t += stride) dst[t] = src[t] * sc;
}

// ------------------------- identity matrix ----------------------------------
__global__ void ident_kernel(float* __restrict__ Z, int n) {
  long idx = (long)blockIdx.x * blockDim.x + threadIdx.x;
  long total = (long)n * n;
  long stride = (long)gridDim.x * blockDim.x;
  for (long t = idx; t < total; t += stride) {
    const int r = (int)(t / n), c = (int)(t % n);
    Z[t] = (r == c) ? 1.f : 0.f;
  }
}

// ------------------------- tiny scalar-prep kernels -------------------------
__global__ void prep_alpha_kernel(float* scal) {
  scal[1] = 1.0f / (scal[0] * scal[0] + 1e-30f);
}
__global__ void prep_nf_kernel(float* scal) {
  scal[4] = sqrtf(scal[3]);                               // nf = ||B||_F
  scal[5] = (scal[4] > 0.f) ? 1.0f / scal[4] : 0.f;       // 1/nf
}
__global__ void final_kernel(float* __restrict__ out, const float* __restrict__ scal) {
  out[0] = scal[2] + scal[7] - 2.0f * sqrtf(scal[4]) * scal[6];
}

// ---------------------------------------------------------------------------
extern "C" void kernel_launch(void* const* d_in, const int* in_sizes, int n_in,
                              void* d_out, int out_size, void* d_ws, size_t ws_size,
                              hipStream_t stream) {
  (void)in_sizes; (void)n_in; (void)out_size; (void)ws_size;
  const int V = 8192, E = 262144, K = 512;

  const float* inputs  = (const float*)d_in[0];  // [V,128]
  const float* targets = (const float*)d_in[1];  // [K,K] SPD
  const int*   eidx    = (const int*)d_in[2];    // [2,E]
  const float* pmat    = (const float*)d_in[3];  // [K,V]
  float* out = (float*)d_out;

  // ---- workspace layout (floats) ----
  float* ws   = (float*)d_ws;
  float* wbuf = ws;                              // E
  float* deg  = wbuf + E;                        // V
  float* scal = deg + V;                         // 64 scalar slots
  float* Yacc = scal + 64;                       // V*K
  float* Mbuf = Yacc + (size_t)V * K;            // V*K
  float* Aadj = Mbuf + (size_t)V * K;            // K*K
  float* X0   = Aadj + (size_t)K * K;
  float* X1   = X0 + (size_t)K * K;
  float* Tm   = X1 + (size_t)K * K;
  float* Bm   = Tm + (size_t)K * K;
  float* Y0   = Bm + (size_t)K * K;
  float* Y1   = Y0 + (size_t)K * K;
  float* Z0   = Y1 + (size_t)K * K;
  float* Z1   = Z0 + (size_t)K * K;

  auto gemm = [&](float* C, const float* A, const float* B, int Mm, int Nn, int Kk,
                  float alpha, float gamma) {
    dim3 grid(Nn / 64, Mm / 64);
    gemm_wmma_f32<<<grid, dim3(128), 0, stream>>>(C, A, B, Mm, Nn, Kk, alpha, gamma);
  };

  // 1) init scratch
  init_kernel<<<1024, 256, 0, stream>>>(Yacc, deg, scal, (long)V * K, V);

  // 2) edge weights (wave per edge)
  edge_weight_kernel<<<E / 8, 256, 0, stream>>>(inputs, eidx, wbuf, E);

  // 3) symmetric scatter into Yacc + degree
  scatter_kernel<<<E, K, 0, stream>>>(wbuf, eidx, pmat, Yacc, deg, E, V, K);

  // 4) M = L @ P^T  (implicit Laplacian)
  form_M_kernel<<<2048, 256, 0, stream>>>(Mbuf, Yacc, deg, pmat, V, K);

  // 5) L_adj = P @ M + 1e-5 I   (big WMMA GEMM: 512 x 8192 x 512)
  gemm(Aadj, pmat, Mbuf, K, K, V, 1.0f, 1e-5f);

  // 6) Newton-Schulz inverse: X <- X (2I - A X)
  rowabsmax_kernel<<<1, 512, 0, stream>>>(Aadj, K, &scal[0]);
  prep_alpha_kernel<<<1, 1, 0, stream>>>(scal);
  scale_kernel<<<512, 256, 0, stream>>>(X0, Aadj, &scal[1], (long)K * K);
  float* Xc = X0; float* Xn = X1;
  const int NS_INV = 24;
  for (int it = 0; it < NS_INV; ++it) {
    gemm(Tm, Aadj, Xc, K, K, K, -1.0f, 2.0f);   // T = 2I - A X
    gemm(Xn, Xc, Tm, K, K, K, 1.0f, 0.0f);      // X = X T
    float* t = Xc; Xc = Xn; Xn = t;
  }
  trace_kernel<<<1, 512, 0, stream>>>(Xc, K, K, &scal[2]);  // tr(S1)

  // 7) B = S0 @ S1 ; tr(sqrtm(B)) via coupled Newton-Schulz on B/||B||_F
  gemm(Bm, targets, Xc, K, K, K, 1.0f, 0.0f);
  fro2_kernel<<<1, 1024, 0, stream>>>(Bm, (long)K * K, &scal[3]);
  prep_nf_kernel<<<1, 1, 0, stream>>>(scal);
  scale_kernel<<<512, 256, 0, stream>>>(Y0, Bm, &scal[5], (long)K * K);
  ident_kernel<<<512, 256, 0, stream>>>(Z0, K);
  float* Yc = Y0; float* Yn = Y1; float* Zc = Z0; float* Zn = Z1;
  const int NS_SQRT = 14;
  for (int it = 0; it < NS_SQRT; ++it) {
    gemm(Tm, Zc, Yc, K, K, K, -0.5f, 1.5f);     // T = (3I - Z Y)/2
    gemm(Yn, Yc, Tm, K, K, K, 1.0f, 0.0f);      // Y = Y T
    gemm(Zn, Tm, Zc, K, K, K, 1.0f, 0.0f);      // Z = T Z
    float* t = Yc; Yc = Yn; Yn = t;
    t = Zc; Zc = Zn; Zn = t;
  }
  trace_kernel<<<1, 512, 0, stream>>>(Yc, K, K, &scal[6]);      // tr(Y_inf)
  trace_kernel<<<1, 512, 0, stream>>>(targets, K, K, &scal[7]); // tr(S0)

  // 8) combine
  final_kernel<<<1, 1, 0, stream>>>(out, scal);
}